// TGCNCell_7215545057454
// MI455X (gfx1250) — compile-verified
//
#include <hip/hip_runtime.h>

typedef __attribute__((ext_vector_type(2))) float v2f;
typedef __attribute__((ext_vector_type(8))) float v8f;

// ---------------------------------------------------------------------------
// degree count: deg[n] = #edges with dst==n   (then deg+2.0 self loop)
// ---------------------------------------------------------------------------
__global__ void k_deg_count(const int* __restrict__ dst,
                            unsigned int* __restrict__ cnt, int E) {
  int e = blockIdx.x * blockDim.x + threadIdx.x;
  if (e < E) atomicAdd(&cnt[dst[e]], 1u);
}

__global__ void k_dinv(const unsigned int* __restrict__ cnt,
                       float* __restrict__ dinv, int n) {
  int i = blockIdx.x * blockDim.x + threadIdx.x;
  if (i < n) dinv[i] = rsqrtf((float)cnt[i] + 2.0f);
}

// ---------------------------------------------------------------------------
// out[M,NOUT] = [A1 | A2] @ W     (A1,A2: [M,64] row-major, W: [128,NOUT])
//
// W is staged in LDS (row stride NOUT+8 floats: 2*(NOUT+8) % 64 == 16, so the
// two half-waves -- which read K-rows k and k+2 -- hit disjoint 16-bank
// groups of the 64 LDS banks; conflict-free ds_loads).
//
// One wave owns a 16-row stripe and ALL NOUT columns: NT = NOUT/16 parallel
// 16x16 accumulators, so each A float2 load feeds NT WMMAs.
//
// V_WMMA_F32_16X16X4_F32 fragment layout (ISA 7.12.2, 32-bit):
//   A 16x4: lanes 0-15 feed K={k0,k0+1} of row M=lane; lanes 16-31 K={k0+2,3}
//   B 4x16: row-striped across lanes per VGPR (mirrors A's K split)
//   C/D:    VGPR j -> row j (lanes 0-15) / j+8 (lanes 16-31), col = lane&15
// ---------------------------------------------------------------------------
template <int NOUT>
__global__ void k_gemm_cat_wmma(const float* __restrict__ A1,
                                const float* __restrict__ A2,
                                const float* __restrict__ W,
                                float* __restrict__ out, int M) {
  constexpr int NT = NOUT / 16;       // column tiles per wave
  constexpr int STRIDE = NOUT + 8;    // padded LDS row stride (floats)
  extern __shared__ float sW[];       // 128 * STRIDE floats (dynamic LDS)

  // cooperative load W[128,NOUT] -> LDS (float4 granules)
  constexpr int QPR = NOUT / 4;       // quads per row
  for (int q = threadIdx.x; q < 128 * QPR; q += blockDim.x) {
    const int k = q / QPR;
    const int c = (q - k * QPR) * 4;
    const float4 w4 = *reinterpret_cast<const float4*>(W + (size_t)k * NOUT + c);
    *reinterpret_cast<float4*>(&sW[k * STRIDE + c]) = w4;
  }
  __syncthreads();

  const int lane = threadIdx.x & 31;
  const int wid  = threadIdx.x >> 5;
  const int mt   = blockIdx.x * (blockDim.x >> 5) + wid;  // 16-row stripe id
  if (mt * 16 >= M) return;            // wave-uniform -> EXEC all-1s for WMMA
  const int hi  = lane >> 4;           // 0: lanes 0-15, 1: lanes 16-31
  const int l   = lane & 15;
  const int kh  = hi * 2;              // this half-wave's K sub-offset
  const int row = mt * 16 + l;

  const float* ap1 = A1 + (size_t)row * 64 + kh;   // x   half of the concat
  const float* ap2 = A2 + (size_t)row * 64 + kh;   // h / r*h half
  const int lbase  = kh * STRIDE + l;              // per-lane LDS base (floats)

  v8f acc[NT];
#pragma unroll
  for (int t = 0; t < NT; ++t) acc[t] = (v8f){};

  // K = 0..63 from A1, K = 64..127 from A2 (concat boundary never straddled:
  // each lane's pair {ka, ka+1} lies inside one 4-wide K step)
#pragma unroll 4
  for (int k0 = 0; k0 < 64; k0 += 4) {
    const float2 av = *reinterpret_cast<const float2*>(ap1 + k0);
    v2f a; a.x = av.x; a.y = av.y;
#pragma unroll
    for (int t = 0; t < NT; ++t) {
      v2f b;
      b.x = sW[lbase + k0 * STRIDE + t * 16];
      b.y = sW[lbase + (k0 + 1) * STRIDE + t * 16];
      acc[t] = __builtin_amdgcn_wmma_f32_16x16x4_f32(
          false, a, false, b, (short)0, acc[t], false, false);
    }
  }
#pragma unroll 4
  for (int k0 = 0; k0 < 64; k0 += 4) {
    const float2 av = *reinterpret_cast<const float2*>(ap2 + k0);
    v2f a; a.x = av.x; a.y = av.y;
#pragma unroll
    for (int t = 0; t < NT; ++t) {
      v2f b;
      b.x = sW[lbase + (64 + k0) * STRIDE + t * 16];
      b.y = sW[lbase + (64 + k0 + 1) * STRIDE + t * 16];
      acc[t] = __builtin_amdgcn_wmma_f32_16x16x4_f32(
          false, a, false, b, (short)0, acc[t], false, false);
    }
  }

  const int rbase = mt * 16 + hi * 8;
#pragma unroll
  for (int t = 0; t < NT; ++t)
#pragma unroll
    for (int j = 0; j < 8; ++j)
      out[(size_t)(rbase + j) * NOUT + t * 16 + l] = acc[t][j];
}

// ---------------------------------------------------------------------------
// agg[n,f] = xw[n,f] * (2*dinv[n]^2) + bias[f]   (self-loop term + bias)
// ---------------------------------------------------------------------------
__global__ void k_init_agg(const float* __restrict__ xw,
                           const float* __restrict__ dinv,
                           const float* __restrict__ bias,
                           float* __restrict__ agg, int n, int fshift) {
  int idx = blockIdx.x * blockDim.x + threadIdx.x;
  if (idx < (n << fshift)) {
    int row = idx >> fshift;
    int f   = idx & ((1 << fshift) - 1);
    float di = dinv[row];
    agg[idx] = xw[idx] * (2.0f * di * di) + bias[f];
  }
}

// ---------------------------------------------------------------------------
// agg[dst] += xw[src] * dinv[src]*dinv[dst]; 1 thread per (edge, 4 features)
// ---------------------------------------------------------------------------
__global__ void k_scatter(const float* __restrict__ xw,
                          const int* __restrict__ src,
                          const int* __restrict__ dst,
                          const float* __restrict__ dinv,
                          float* __restrict__ agg, int E, int fshift) {
  const int qshift = fshift - 2;                    // quads per edge = F/4
  unsigned int tid   = blockIdx.x * blockDim.x + threadIdx.x;
  unsigned int total = (unsigned int)E << qshift;
  if (tid >= total) return;
  int e = tid >> qshift;
  int f = (tid & ((1u << qshift) - 1u)) << 2;
  int s = src[e], d = dst[e];
  float norm = dinv[s] * dinv[d];
  const float4 v = *reinterpret_cast<const float4*>(xw + ((size_t)s << fshift) + f);
  float* o = agg + ((size_t)d << fshift) + f;
  atomicAdd(o + 0, v.x * norm);
  atomicAdd(o + 1, v.y * norm);
  atomicAdd(o + 2, v.z * norm);
  atomicAdd(o + 3, v.w * norm);
}

__global__ void k_sigmoid(float* __restrict__ a, int n) {
  int i = blockIdx.x * blockDim.x + threadIdx.x;
  if (i < n) a[i] = 1.0f / (1.0f + expf(-a[i]));
}

// r = first N*64 flat elements of ru (faithful torch reshape, B=1)
__global__ void k_rh(const float* __restrict__ ru, const float* __restrict__ h,
                     float* __restrict__ rh, int n64) {
  int i = blockIdx.x * blockDim.x + threadIdx.x;
  if (i < n64) rh[i] = ru[i] * h[i];
}

// u = second N*64 flat elements; h' = u*h + (1-u)*tanh(agg2)
__global__ void k_final(const float* __restrict__ agg2, const float* __restrict__ ru,
                        const float* __restrict__ h, float* __restrict__ out, int n64) {
  int i = blockIdx.x * blockDim.x + threadIdx.x;
  if (i < n64) {
    float c = tanhf(agg2[i]);
    float u = ru[n64 + i];
    out[i] = u * h[i] + (1.0f - u) * c;
  }
}

// ---------------------------------------------------------------------------
extern "C" void kernel_launch(void* const* d_in, const int* in_sizes, int n_in,
                              void* d_out, int out_size, void* d_ws, size_t ws_size,
                              hipStream_t stream) {
  const float* x  = (const float*)d_in[0];     // [N,64]
  const int*   ei = (const int*)d_in[1];       // [2,E]: src row 0, dst row 1
  const float* h  = (const float*)d_in[3];     // [N,64]
  const float* W1 = (const float*)d_in[4];     // [128,128]
  const float* b1 = (const float*)d_in[5];     // [128]
  const float* W2 = (const float*)d_in[6];     // [128,64]
  const float* b2 = (const float*)d_in[7];     // [64]
  float* out = (float*)d_out;                  // [N,64]

  const int N = in_sizes[0] / 64;              // 50000
  const int E = in_sizes[2];                   // edge_weight length == E
  const int* src = ei;
  const int* dst = ei + E;

  // workspace carve-up (450*N floats = ~90 MB; whole pipeline is L2-resident)
  float* ws = (float*)d_ws;
  unsigned int* cnt = (unsigned int*)ws;                 // N u32
  float* dinv = ws + (size_t)N;                          // N
  float* xw1  = ws + (size_t)2 * N;                      // N*128
  float* ru   = xw1 + (size_t)N * 128;                   // N*128 (agg1 -> sigmoid in place)
  float* rh   = ru  + (size_t)N * 128;                   // N*64
  float* xw2  = rh  + (size_t)N * 64;                    // N*64
  float* agg2 = xw2 + (size_t)N * 64;                    // N*64

  hipMemsetAsync(cnt, 0, sizeof(unsigned int) * (size_t)N, stream);

  const int B = 256;
  k_deg_count<<<(E + B - 1) / B, B, 0, stream>>>(dst, cnt, E);
  k_dinv<<<(N + B - 1) / B, B, 0, stream>>>(cnt, dinv, N);

  const int mtiles = (N + 15) / 16;            // 3125 row stripes, 8 waves/block

  // --- GCN 1: [x|h] @ W1 -> xw1 [N,128] ---
  {
    const size_t lds = 128 * (128 + 8) * sizeof(float);  // 69,632 B
    k_gemm_cat_wmma<128><<<(mtiles + 7) / 8, 256, lds, stream>>>(x, h, W1, xw1, N);
  }
  k_init_agg<<<((N * 128) + B - 1) / B, B, 0, stream>>>(xw1, dinv, b1, ru, N, 7);
  k_scatter<<<((E * 32) + B - 1) / B, B, 0, stream>>>(xw1, src, dst, dinv, ru, E, 7);
  k_sigmoid<<<((N * 128) + B - 1) / B, B, 0, stream>>>(ru, N * 128);
  k_rh<<<((N * 64) + B - 1) / B, B, 0, stream>>>(ru, h, rh, N * 64);

  // --- GCN 2: [x|r*h] @ W2 -> xw2 [N,64] ---
  {
    const size_t lds = 128 * (64 + 8) * sizeof(float);   // 36,864 B
    k_gemm_cat_wmma<64><<<(mtiles + 7) / 8, 256, lds, stream>>>(x, rh, W2, xw2, N);
  }
  k_init_agg<<<((N * 64) + B - 1) / B, B, 0, stream>>>(xw2, dinv, b2, agg2, N, 6);
  k_scatter<<<((E * 16) + B - 1) / B, B, 0, stream>>>(xw2, src, dst, dinv, agg2, E, 6);
  k_final<<<((N * 64) + B - 1) / B, B, 0, stream>>>(agg2, ru, h, out, N * 64);
}